// GraphAttentionNetwork_87411174408212
// MI455X (gfx1250) — compile-verified
//
#include <hip/hip_runtime.h>
#include <hip/hip_bf16.h>

// ---------------------------------------------------------------------------
// GAT on gfx1250. Dense [N,N] softmax collapses to sparse CSR math because the
// fill constant is +-9e15 (see analysis). GEMMs use V_WMMA_F32_16X16X4_F32
// with the weight slice staged in LDS via GLOBAL_LOAD_ASYNC_TO_LDS_B128 and
// register double-buffering of the fragments.
// ---------------------------------------------------------------------------

typedef __attribute__((ext_vector_type(2))) float v2f;
typedef __attribute__((ext_vector_type(8))) float v8f;

#define GAT_N 8192
#define GAT_E 262144

#if defined(__HIP_DEVICE_COMPILE__) && \
    __has_builtin(__builtin_amdgcn_global_load_async_to_lds_b128) && \
    __has_builtin(__builtin_amdgcn_s_wait_asynccnt)
#define HAVE_ASYNC_LDS 1
typedef int v4i_vec __attribute__((vector_size(4 * sizeof(int))));
typedef __attribute__((address_space(1))) v4i_vec* as1_v4i_ptr;
typedef __attribute__((address_space(3))) v4i_vec* as3_v4i_ptr;
#else
#define HAVE_ASYNC_LDS 0
#endif

// ---------------- utility kernels ----------------

__global__ void k_zero_u32(unsigned int* p, int n) {
  int i = blockIdx.x * blockDim.x + threadIdx.x;
  if (i < n) p[i] = 0u;
}

// flags[k] = sum(a1[k,:]) for k=0..3 ; flags[4] = sum(a2)
__global__ void k_flags(const float* __restrict__ a1, const float* __restrict__ a2,
                        float* __restrict__ flags) {
  __shared__ float red[256];
  int t = threadIdx.x;
  for (int k = 0; k < 4; ++k) {
    float v = (t < 128) ? a1[k * 128 + t] : 0.f;
    red[t] = v;
    __syncthreads();
    for (int off = 128; off > 0; off >>= 1) {
      if (t < off) red[t] += red[t + off];
      __syncthreads();
    }
    if (t == 0) flags[k] = red[0];
    __syncthreads();
  }
  red[t] = a2[t];  // 256 entries
  __syncthreads();
  for (int off = 128; off > 0; off >>= 1) {
    if (t < off) red[t] += red[t + off];
    __syncthreads();
  }
  if (t == 0) flags[4] = red[0];
}

// ---------------- CSR build (by src) ----------------

__global__ void k_deg(const int* __restrict__ src, int* __restrict__ deg, int E) {
  int e = blockIdx.x * blockDim.x + threadIdx.x;
  if (e < E) atomicAdd(&deg[src[e]], 1);
}

// Single block, 1024 threads, 8 elements/thread: exclusive scan of deg[0..n)
__global__ void k_scan(const int* __restrict__ deg, int* __restrict__ rowptr,
                       int* __restrict__ cursor, int n) {
  __shared__ int sums[1024];
  int t = threadIdx.x;
  int base = t * 8;
  int local[8];
  int s = 0;
  for (int i = 0; i < 8; ++i) {
    int v = (base + i < n) ? deg[base + i] : 0;
    s += v;
    local[i] = s;  // inclusive within thread
  }
  sums[t] = s;
  __syncthreads();
  for (int off = 1; off < 1024; off <<= 1) {
    int v = (t >= off) ? sums[t - off] : 0;
    __syncthreads();
    sums[t] += v;
    __syncthreads();
  }
  int excl = (t == 0) ? 0 : sums[t - 1];
  if (t == 0) rowptr[0] = 0;
  for (int i = 0; i < 8; ++i) {
    if (base + i < n) {
      rowptr[base + i + 1] = excl + local[i];
      cursor[base + i] = excl + (i ? local[i - 1] : 0);
    }
  }
}

__global__ void k_scatter(const int* __restrict__ src, const int* __restrict__ dst,
                          int* __restrict__ cursor, int* __restrict__ colv, int E) {
  int e = blockIdx.x * blockDim.x + threadIdx.x;
  if (e < E) {
    int p = atomicAdd(&cursor[src[e]], 1);
    colv[p] = dst[e];
  }
}

// ---------------- WMMA f32 GEMM:  C[M,Nc] = A[M,K] @ B^T, B is [Nc,K] --------
// Block = 8 waves = 128 rows x 64 cols. The 64-row B slice is staged in LDS
// (stride 260 floats -> fragment ds_load_b64 is bank-conflict-free), loaded
// with GLOBAL_LOAD_ASYNC_TO_LDS_B128 when the builtin exists. Each wave runs a
// 16x64 tile: 4 accumulators, K-loop step 4 of V_WMMA_F32_16X16X4_F32 with
// register double-buffered fragments so loads overlap the 4 in-flight WMMAs.
// A-frag layout (ISA 7.12.2, 32-bit A 16x4): lanes 0-15 hold M=0..15 with
// K=kk,kk+1; lanes 16-31 the same rows with K=kk+2,kk+3. B-frag is the
// transpose-symmetric layout; C/D: VGPR v holds M=v (+8 upper half), lane&15=N.

#define BSTRIDE 260  // floats per staged B row (256 + 4 pad)

__global__ void k_gemm_wmma(const float* __restrict__ A, const float* __restrict__ B,
                            float* __restrict__ C, int M, int K, int Nc) {
  __shared__ float Bl[64 * BSTRIDE];  // 66560 B
  int t = threadIdx.x;
  int wave = t >> 5;
  int lane = t & 31;
  int m0 = blockIdx.x * 128 + wave * 16;
  int n0 = blockIdx.y * 64;

  // ---- stage B slice: 64 rows x K floats (K == 256 here) ----
  {
    int nquads = (64 * K) >> 2;            // float4 count = 4096
    for (int linear = t; linear < nquads; linear += 256) {
      int row = linear >> 6;               // K/4 == 64 quads per row
      int q = linear & 63;
      const float* gsrc = B + (size_t)(n0 + row) * K + q * 4;
      float* ldst = &Bl[row * BSTRIDE + q * 4];
#if HAVE_ASYNC_LDS
      __builtin_amdgcn_global_load_async_to_lds_b128(
          (as1_v4i_ptr)(uintptr_t)gsrc,
          (as3_v4i_ptr)(unsigned)(uintptr_t)ldst, 0, 0);
#else
      *(float4*)ldst = *(const float4*)gsrc;
#endif
    }
#if HAVE_ASYNC_LDS
    __builtin_amdgcn_s_wait_asynccnt(0);
#endif
    __syncthreads();
  }

  int r = lane & 15;
  int ko = (lane >> 4) << 1;  // 0 or 2
  const float* Arow = A + (size_t)(m0 + r) * K;
  int bidx0 = (0 * 16 + r) * BSTRIDE + ko;
  int bidx1 = (1 * 16 + r) * BSTRIDE + ko;
  int bidx2 = (2 * 16 + r) * BSTRIDE + ko;
  int bidx3 = (3 * 16 + r) * BSTRIDE + ko;

  v8f acc0 = {}, acc1 = {}, acc2 = {}, acc3 = {};

  // prologue fragments (k-step 0)
  v2f a_c, b0c, b1c, b2c, b3c;
  {
    float2 av = *(const float2*)(Arow + ko);
    a_c.x = av.x; a_c.y = av.y;
    float2 f0 = *(const float2*)&Bl[bidx0];
    float2 f1 = *(const float2*)&Bl[bidx1];
    float2 f2 = *(const float2*)&Bl[bidx2];
    float2 f3 = *(const float2*)&Bl[bidx3];
    b0c.x = f0.x; b0c.y = f0.y;
    b1c.x = f1.x; b1c.y = f1.y;
    b2c.x = f2.x; b2c.y = f2.y;
    b3c.x = f3.x; b3c.y = f3.y;
  }

  for (int kk = 0; kk < K; kk += 4) {
    int kn = kk + 4;
    if (kn >= K) kn = 0;  // clamped prefetch (last result discarded)
    // issue next-step loads before this step's WMMAs
    float2 av = *(const float2*)(Arow + kn + ko);
    float2 f0 = *(const float2*)&Bl[bidx0 + kn];
    float2 f1 = *(const float2*)&Bl[bidx1 + kn];
    float2 f2 = *(const float2*)&Bl[bidx2 + kn];
    float2 f3 = *(const float2*)&Bl[bidx3 + kn];

    acc0 = __builtin_amdgcn_wmma_f32_16x16x4_f32(false, a_c, false, b0c, (short)0, acc0, false, false);
    acc1 = __builtin_amdgcn_wmma_f32_16x16x4_f32(false, a_c, false, b1c, (short)0, acc1, false, false);
    acc2 = __builtin_amdgcn_wmma_f32_16x16x4_f32(false, a_c, false, b2c, (short)0, acc2, false, false);
    acc3 = __builtin_amdgcn_wmma_f32_16x16x4_f32(false, a_c, false, b3c, (short)0, acc3, false, false);

    a_c.x = av.x; a_c.y = av.y;
    b0c.x = f0.x; b0c.y = f0.y;
    b1c.x = f1.x; b1c.y = f1.y;
    b2c.x = f2.x; b2c.y = f2.y;
    b3c.x = f3.x; b3c.y = f3.y;
  }

  int mrow = m0 + ((lane >> 4) << 3);  // +8 for upper half-wave
  for (int v = 0; v < 8; ++v) {
    size_t row = (size_t)(mrow + v) * Nc;
    C[row + n0 + 0 * 16 + r] = acc0[v];
    C[row + n0 + 1 * 16 + r] = acc1[v];
    C[row + n0 + 2 * 16 + r] = acc2[v];
    C[row + n0 + 3 * 16 + r] = acc3[v];
  }
}

// ---------------- per-(node,head) score dots ----------------
// ssrc[k*N+n] = dot(Wh[n, k*D : k*D+D], a[k][0:D]); sdst likewise with a[k][D:2D]

__global__ void k_scores(const float* __restrict__ Wh, const float* __restrict__ avec,
                         float* __restrict__ ssrc, float* __restrict__ sdst,
                         int N, int D, int nHeads, int stride) {
  int wv = (blockIdx.x * blockDim.x + threadIdx.x) >> 5;
  int lane = threadIdx.x & 31;
  if (wv >= N * nHeads) return;
  int n = wv % N;
  int k = wv / N;
  const float* row = Wh + (size_t)n * stride + k * D;
  const float* as = avec + k * (2 * D);
  const float* ad = as + D;
  float d1 = 0.f, d2 = 0.f;
  for (int d = lane; d < D; d += 32) {
    float w = row[d];
    d1 += w * as[d];
    d2 += w * ad[d];
  }
  for (int off = 16; off > 0; off >>= 1) {
    d1 += __shfl_xor(d1, off, 32);
    d2 += __shfl_xor(d2, off, 32);
  }
  if (lane == 0) {
    ssrc[k * N + n] = d1;
    sdst[k * N + n] = d2;
  }
}

// ---------------- column sums:  S[c] = sum_n Wh[n,c] ----------------

__global__ void k_colsum(const float* __restrict__ Wh, float* __restrict__ S,
                         int rows, int cols, int rowsPerBlock) {
  int c = threadIdx.x;   // blockDim.x == cols
  int r0 = blockIdx.x * rowsPerBlock;
  int r1 = r0 + rowsPerBlock;
  if (r1 > rows) r1 = rows;
  float acc = 0.f;
  for (int r = r0; r < r1; ++r) acc += Wh[(size_t)r * cols + c];
  atomicAdd(&S[c], acc);
}

// ---------------- sparse aggregation + ELU ----------------
// One wave per (node, head). Regime picked by sign(sum(a)).

__global__ void k_agg(const float* __restrict__ Wh, int stride, int D, int nHeads,
                      const float* __restrict__ ssrc, const float* __restrict__ sdst,
                      const float* __restrict__ S, const float* __restrict__ flags,
                      int flagBase, const int* __restrict__ rowptr,
                      const int* __restrict__ colv, float* __restrict__ out,
                      int outStride, int eluCount, int N) {
  int wv = (blockIdx.x * blockDim.x + threadIdx.x) >> 5;
  int lane = threadIdx.x & 31;
  if (wv >= N * nHeads) return;
  int n = wv % N;
  int k = wv / N;
  float sa = flags[flagBase + k];
  int start = rowptr[n];
  int end = rowptr[n + 1];
  int deg = end - start;
  int nd = D >> 5;                  // dims per lane: 2 (D=64) or 4 (D=128)
  size_t colOff = (size_t)k * D;
  const float* Scol = S + k * D;
  float acc[4] = {0.f, 0.f, 0.f, 0.f};
  float outv[4];

  if (sa > 0.f) {
    // fill = -9e15*sum(a) is hugely negative -> softmax over edges only
    if (deg == 0) {
      for (int i = 0; i < nd; ++i) outv[i] = Scol[lane + 32 * i] / (float)N;
    } else {
      float si = ssrc[k * N + n];
      float m = -3.4e38f;
      for (int e = start + lane; e < end; e += 32) {
        float sc = si + sdst[k * N + colv[e]];
        sc = sc > 0.f ? sc : 0.2f * sc;          // leaky_relu(0.2)
        m = fmaxf(m, sc);
      }
      for (int off = 16; off > 0; off >>= 1) m = fmaxf(m, __shfl_xor(m, off, 32));
      float denom = 0.f;
      for (int e = start; e < end; ++e) {
        int j = colv[e];
        float sc = si + sdst[k * N + j];
        sc = sc > 0.f ? sc : 0.2f * sc;
        float w = __expf(sc - m);
        denom += w;
        const float* wr = Wh + (size_t)j * stride + colOff;
        for (int i = 0; i < nd; ++i) acc[i] += w * wr[lane + 32 * i];
      }
      float inv = 1.f / denom;
      for (int i = 0; i < nd; ++i) outv[i] = acc[i] * inv;
    }
  } else {
    // fill hugely positive -> uniform attention over NON-neighbors
    for (int e = start; e < end; ++e) {
      const float* wr = Wh + (size_t)colv[e] * stride + colOff;
      for (int i = 0; i < nd; ++i) acc[i] += wr[lane + 32 * i];
    }
    float inv = 1.f / (float)(N - deg);
    for (int i = 0; i < nd; ++i) outv[i] = (Scol[lane + 32 * i] - acc[i]) * inv;
  }

  float* orow = out + (size_t)n * outStride + colOff;
  for (int i = 0; i < nd; ++i) {
    float v = outv[i];
    for (int r = 0; r < eluCount; ++r) v = v > 0.f ? v : (__expf(v) - 1.f);
    orow[lane + 32 * i] = v;
  }
}

// ---------------- launcher ----------------

extern "C" void kernel_launch(void* const* d_in, const int* in_sizes, int n_in,
                              void* d_out, int out_size, void* d_ws, size_t ws_size,
                              hipStream_t stream) {
  const float* h  = (const float*)d_in[0];   // [N,256]
  const float* W1 = (const float*)d_in[1];   // [4,64,256] == [256,256] row-major
  const float* a1 = (const float*)d_in[2];   // [4,128]
  const float* W2 = (const float*)d_in[3];   // [128,256]
  const float* a2 = (const float*)d_in[4];   // [256]
  const int* esrc = (const int*)d_in[5];     // [E]
  const int* edst = (const int*)d_in[6];     // [E]
  float* out = (float*)d_out;                // [N,128]

  const int N = GAT_N;
  const int E = GAT_E;

  // workspace carve-up (256B aligned slabs)
  char* ws = (char*)d_ws;
  size_t off = 0;
  auto take = [&](size_t bytes) {
    char* p = ws + off;
    off += (bytes + 255) & ~(size_t)255;
    return p;
  };
  float* Wh1   = (float*)take((size_t)N * 256 * 4);
  float* X     = (float*)take((size_t)N * 256 * 4);
  float* Wh2   = (float*)take((size_t)N * 128 * 4);
  float* s1src = (float*)take((size_t)4 * N * 4);
  float* s1dst = (float*)take((size_t)4 * N * 4);
  float* s2src = (float*)take((size_t)N * 4);
  float* s2dst = (float*)take((size_t)N * 4);
  float* S1    = (float*)take(256 * 4);
  float* S2    = (float*)take(128 * 4);
  float* flags = (float*)take(8 * 4);
  int* deg     = (int*)take((size_t)N * 4);
  int* rowptr  = (int*)take((size_t)(N + 1) * 4);
  int* cursor  = (int*)take((size_t)N * 4);
  int* colv    = (int*)take((size_t)E * 4);
  (void)ws_size; (void)in_sizes; (void)n_in; (void)out_size;

  // 0) zero accumulators
  k_zero_u32<<<1, 256, 0, stream>>>((unsigned*)S1, 256);
  k_zero_u32<<<1, 128, 0, stream>>>((unsigned*)S2, 128);
  k_zero_u32<<<N / 256, 256, 0, stream>>>((unsigned*)deg, N);

  // 1) per-head sum(a) flags
  k_flags<<<1, 256, 0, stream>>>(a1, a2, flags);

  // 2) CSR by src
  k_deg<<<E / 256, 256, 0, stream>>>(esrc, deg, E);
  k_scan<<<1, 1024, 0, stream>>>(deg, rowptr, cursor, N);
  k_scatter<<<E / 256, 256, 0, stream>>>(esrc, edst, cursor, colv, E);

  // ---- layer 1 ----
  // Wh1[N,256] = h @ W1^T  : grid 64 m-blocks x 4 n-blocks, 8 waves each
  k_gemm_wmma<<<dim3(N / 128, 256 / 64), 256, 0, stream>>>(h, W1, Wh1, N, 256, 256);
  // scores: N*4 waves
  k_scores<<<(N * 4) / 8, 256, 0, stream>>>(Wh1, a1, s1src, s1dst, N, 64, 4, 256);
  // column sums
  k_colsum<<<N / 64, 256, 0, stream>>>(Wh1, S1, N, 256, 64);
  // aggregate + elu -> X[N,256]
  k_agg<<<(N * 4) / 8, 256, 0, stream>>>(Wh1, 256, 64, 4, s1src, s1dst, S1, flags, 0,
                                         rowptr, colv, X, 256, 1, N);

  // ---- layer 2 ----
  // Wh2[N,128] = X @ W2^T : grid 64 x 2
  k_gemm_wmma<<<dim3(N / 128, 128 / 64), 256, 0, stream>>>(X, W2, Wh2, N, 256, 128);
  k_scores<<<N / 8, 256, 0, stream>>>(Wh2, a2, s2src, s2dst, N, 128, 1, 128);
  k_colsum<<<N / 64, 128, 0, stream>>>(Wh2, S2, N, 128, 64);
  // elu applied twice (head elu + final elu) -> d_out[N,128]
  k_agg<<<N / 8, 256, 0, stream>>>(Wh2, 128, 128, 1, s2src, s2dst, S2, flags, 4,
                                   rowptr, colv, out, 128, 2, N);
}